// GCN_Xplain_v8_ap_27023934227208
// MI455X (gfx1250) — compile-verified
//
#include <hip/hip_runtime.h>

// ============================================================================
// GNN (LEConv+GCNConv x3 -> mean-pool -> MLP) for MI455X (gfx1250, wave32).
//
// Roofline reasoning:
//  - Dense GEMMs ~38 GFLOP fp32 -> V_WMMA_F32_16X16X4_F32 (fp32 precision of
//    the reference, runs on the matrix pipe).
//  - Edge scatter/gather ~2-3 GB of traffic; every tensor (<=41MB) fits in the
//    192MB L2 -> float4 gathers + global_atomic_add_f32 suffice at 23.3 TB/s.
//  - LEConv's three GEMMs share input X -> fused triple-output WMMA GEMM
//    (one A-tile load feeds 12 WMMAs across 3 weight matrices x 4 col-tiles).
//  - B operand staged TRANSPOSED in LDS, rows padded to 34 floats so each
//    lane's (k,k+1) pair is one aligned ds_load_b64 (no VGPR shuffles before
//    WMMA) and lanes land on distinct banks.
//  - 4 accumulators per wave = 4 independent WMMA dependency chains.
// ============================================================================

typedef __attribute__((ext_vector_type(2))) float v2f;
typedef __attribute__((ext_vector_type(8))) float v8f;

#define NN   20000
#define EE   160000
#define FIN  128
#define HD1  128
#define HD2  256
#define HD3  512
#define GG   64

#define T_ROWS 128   // rows per block (8 waves x 16)
#define T_COLS 64    // cols per block (4 WMMA col-tiles per wave)
#define KCH    32    // K-chunk staged in LDS
#define KPAD   (KCH + 2)   // 34 floats: keeps rows 8B-aligned, avoids bank hits

// ---------------------------------------------------------------------------
// Generic WMMA GEMM: Y[N,M] = X[N,K] @ W[K,M] (+bias) (+ReLU)
// Wave computes a 16x64 strip (4 f32 accumulators). W chunk staged transposed
// in LDS. A-layout (16x4 f32): lane<16 holds rows r0+lane, k..k+1; lane>=16
// same rows, k+2..k+3. B-layout (4x16): mirrored. C/D: VGPR g = row r0+g
// (lanes 16-31: +8).
// ---------------------------------------------------------------------------
__global__ __launch_bounds__(256) void gemm_wmma_f32(
    const float* __restrict__ X, const float* __restrict__ W,
    const float* __restrict__ bias, float* __restrict__ Y,
    int N, int K, int M, int do_relu)
{
  __shared__ float sWt[T_COLS][KPAD];    // transposed chunk: sWt[col][k]
  const int wave = threadIdx.x >> 5;
  const int lane = threadIdx.x & 31;
  const int half = lane >> 4;
  const int l15  = lane & 15;
  const int r0 = blockIdx.x * T_ROWS + wave * 16;
  const int c0 = blockIdx.y * T_COLS;

  int arow = r0 + l15; if (arow > N - 1) arow = N - 1;   // clamp (stores guarded)
  const float* __restrict__ Xr = X + (size_t)arow * K;

  v8f acc[4] = {};
  for (int k0 = 0; k0 < K; k0 += KCH) {
    for (int i = threadIdx.x; i < KCH * T_COLS; i += 256) {
      int kk = i >> 6, cc = i & 63;          // consecutive threads: consecutive cols
      int col = c0 + cc;
      sWt[cc][kk] = (col < M) ? W[(size_t)(k0 + kk) * M + col] : 0.0f;
    }
    __syncthreads();
#pragma unroll
    for (int ks = 0; ks < KCH; ks += 4) {
      v2f a = *(const v2f*)(Xr + k0 + ks + half * 2);
      const int sk = ks + half * 2;
#pragma unroll
      for (int j = 0; j < 4; ++j) {
        v2f b = *(const v2f*)(&sWt[j * 16 + l15][sk]);   // one ds_load_b64
        acc[j] = __builtin_amdgcn_wmma_f32_16x16x4_f32(false, a, false, b,
                                                       (short)0, acc[j], false, false);
      }
    }
    __syncthreads();
  }
#pragma unroll
  for (int g = 0; g < 8; ++g) {
    int row = r0 + g + half * 8;
    if (row < N) {
#pragma unroll
      for (int j = 0; j < 4; ++j) {
        int col = c0 + j * 16 + l15;
        if (col < M) {
          float v = acc[j][g] + (bias ? bias[col] : 0.0f);
          if (do_relu) v = fmaxf(v, 0.0f);
          Y[(size_t)row * M + col] = v;
        }
      }
    }
  }
}

// ---------------------------------------------------------------------------
// Fused LEConv GEMMs: Ya = X@W1+b1, Yb = X@W2, Yc = X@W3+b3.
// One A-tile load feeds 12 WMMAs (3 weight matrices x 4 col-tiles).
// ---------------------------------------------------------------------------
__global__ __launch_bounds__(256) void gemm3_wmma_f32(
    const float* __restrict__ X,
    const float* __restrict__ W1, const float* __restrict__ b1,
    const float* __restrict__ W2,
    const float* __restrict__ W3, const float* __restrict__ b3,
    float* __restrict__ Ya, float* __restrict__ Yb, float* __restrict__ Yc,
    int N, int K, int M)
{
  __shared__ float sWt[3][T_COLS][KPAD];   // ~25.5 KB
  const int wave = threadIdx.x >> 5;
  const int lane = threadIdx.x & 31;
  const int half = lane >> 4;
  const int l15  = lane & 15;
  const int r0 = blockIdx.x * T_ROWS + wave * 16;
  const int c0 = blockIdx.y * T_COLS;

  int arow = r0 + l15; if (arow > N - 1) arow = N - 1;
  const float* __restrict__ Xr = X + (size_t)arow * K;

  v8f acc[3][4] = {};
  for (int k0 = 0; k0 < K; k0 += KCH) {
    for (int i = threadIdx.x; i < KCH * T_COLS; i += 256) {
      int kk = i >> 6, cc = i & 63;
      int col = c0 + cc;
      bool ok = (col < M);
      size_t widx = (size_t)(k0 + kk) * M + (ok ? col : 0);
      sWt[0][cc][kk] = ok ? W1[widx] : 0.0f;
      sWt[1][cc][kk] = ok ? W2[widx] : 0.0f;
      sWt[2][cc][kk] = ok ? W3[widx] : 0.0f;
    }
    __syncthreads();
#pragma unroll
    for (int ks = 0; ks < KCH; ks += 4) {
      v2f a = *(const v2f*)(Xr + k0 + ks + half * 2);
      const int sk = ks + half * 2;
#pragma unroll
      for (int m = 0; m < 3; ++m) {
#pragma unroll
        for (int j = 0; j < 4; ++j) {
          v2f b = *(const v2f*)(&sWt[m][j * 16 + l15][sk]);
          acc[m][j] = __builtin_amdgcn_wmma_f32_16x16x4_f32(false, a, false, b,
                                      (short)0, acc[m][j], false, false);
        }
      }
    }
    __syncthreads();
  }
#pragma unroll
  for (int g = 0; g < 8; ++g) {
    int row = r0 + g + half * 8;
    if (row < N) {
#pragma unroll
      for (int j = 0; j < 4; ++j) {
        int col = c0 + j * 16 + l15;
        if (col < M) {
          size_t o = (size_t)row * M + col;
          Ya[o] = acc[0][j][g] + b1[col];
          Yb[o] = acc[1][j][g];
          Yc[o] = acc[2][j][g] + b3[col];
        }
      }
    }
  }
}

// ---------------------------------------------------------------------------
// Elementwise / edge kernels
// ---------------------------------------------------------------------------
__global__ void set_val_k(float* __restrict__ p, long long n, float v) {
  long long t = (long long)blockIdx.x * blockDim.x + threadIdx.x;
  if (t < n) p[t] = v;
}

__global__ void relu_k(float* __restrict__ p, long long n) {
  long long t = (long long)blockIdx.x * blockDim.x + threadIdx.x;
  if (t < n) p[t] = fmaxf(p[t], 0.0f);
}

__global__ void deg_edge_k(float* __restrict__ deg, const int* __restrict__ dst,
                           const float* __restrict__ ew, int E) {
  int e = blockIdx.x * blockDim.x + threadIdx.x;
  if (e < E) atomicAdd(&deg[dst[e]], ew[e]);
}

__global__ void dinv_k(const float* __restrict__ deg, float* __restrict__ dinv, int N) {
  int i = blockIdx.x * blockDim.x + threadIdx.x;
  if (i < N) { float d = deg[i]; dinv[i] = (d > 0.0f) ? rsqrtf(d) : 0.0f; }
}

// LEConv edge message: agg[dst] += (a[src] - b[dst]) * ew
__global__ void le_edge_k(const float* __restrict__ A, const float* __restrict__ Bm,
                          const float* __restrict__ ew, const int* __restrict__ src,
                          const int* __restrict__ dst, float* __restrict__ Agg,
                          int E, int F)
{
  long long t = (long long)blockIdx.x * blockDim.x + threadIdx.x;
  int nch = F >> 2;
  if (t >= (long long)E * nch) return;
  int e = (int)(t / nch);
  int f = ((int)(t % nch)) << 2;
  int s = src[e], d = dst[e];
  float w = ew[e];
  const float4 av = *(const float4*)(A  + (size_t)s * F + f);
  const float4 bv = *(const float4*)(Bm + (size_t)d * F + f);
  float* p = Agg + (size_t)d * F + f;
  atomicAdd(p + 0, (av.x - bv.x) * w);
  atomicAdd(p + 1, (av.y - bv.y) * w);
  atomicAdd(p + 2, (av.z - bv.z) * w);
  atomicAdd(p + 3, (av.w - bv.w) * w);
}

// GCN self-loop + bias init: out[i] = xw[i] * dinv[i]^2 + bias
__global__ void gcn_init_k(const float* __restrict__ xw, const float* __restrict__ dinv,
                           const float* __restrict__ bias, float* __restrict__ out,
                           int N, int F)
{
  long long t = (long long)blockIdx.x * blockDim.x + threadIdx.x;
  if (t >= (long long)N * F) return;
  int n = (int)(t / F);
  int f = (int)(t % F);
  float di = dinv[n];
  out[t] = xw[t] * di * di + bias[f];
}

// GCN edge message: out[dst] += xw[src] * (dinv[src]*ew*dinv[dst])
__global__ void gcn_edge_k(const float* __restrict__ xw, const float* __restrict__ dinv,
                           const float* __restrict__ ew, const int* __restrict__ src,
                           const int* __restrict__ dst, float* __restrict__ out,
                           int E, int F)
{
  long long t = (long long)blockIdx.x * blockDim.x + threadIdx.x;
  int nch = F >> 2;
  if (t >= (long long)E * nch) return;
  int e = (int)(t / nch);
  int f = ((int)(t % nch)) << 2;
  int s = src[e], d = dst[e];
  float sc = dinv[s] * ew[e] * dinv[d];
  const float4 xv = *(const float4*)(xw + (size_t)s * F + f);
  float* p = out + (size_t)d * F + f;
  atomicAdd(p + 0, xv.x * sc);
  atomicAdd(p + 1, xv.y * sc);
  atomicAdd(p + 2, xv.z * sc);
  atomicAdd(p + 3, xv.w * sc);
}

// Mean-pool
__global__ void pool_acc_k(const float* __restrict__ H, const int* __restrict__ bat,
                           float* __restrict__ sums, int N, int F)
{
  long long t = (long long)blockIdx.x * blockDim.x + threadIdx.x;
  int nch = F >> 2;
  if (t >= (long long)N * nch) return;
  int n = (int)(t / nch);
  int f = ((int)(t % nch)) << 2;
  int g = bat[n];
  const float4 hv = *(const float4*)(H + (size_t)n * F + f);
  float* p = sums + (size_t)g * F + f;
  atomicAdd(p + 0, hv.x);
  atomicAdd(p + 1, hv.y);
  atomicAdd(p + 2, hv.z);
  atomicAdd(p + 3, hv.w);
}

__global__ void pool_cnt_k(const int* __restrict__ bat, float* __restrict__ cnt, int N) {
  int n = blockIdx.x * blockDim.x + threadIdx.x;
  if (n < N) atomicAdd(&cnt[bat[n]], 1.0f);
}

__global__ void pool_div_k(const float* __restrict__ sums, const float* __restrict__ cnt,
                           float* __restrict__ hg, int G, int F)
{
  int t = blockIdx.x * blockDim.x + threadIdx.x;
  if (t >= G * F) return;
  hg[t] = sums[t] / fmaxf(cnt[t / F], 1.0f);
}

// ---------------------------------------------------------------------------
// Host launcher
// ---------------------------------------------------------------------------
extern "C" void kernel_launch(void* const* d_in, const int* in_sizes, int n_in,
                              void* d_out, int out_size, void* d_ws, size_t ws_size,
                              hipStream_t stream)
{
  (void)in_sizes; (void)n_in; (void)out_size; (void)ws_size;

  const float* x   = (const float*)d_in[0];
  const int*   ei  = (const int*)d_in[1];
  const float* ew  = (const float*)d_in[2];
  const int*   bat = (const int*)d_in[3];
  const int* src = ei;
  const int* dst = ei + EE;

  // per-layer params: base 4 + 7*(i-1): W1,b1,W2,W3,b3,gW,gb
  const float* P[31];
  for (int i = 0; i < 31; ++i) P[i] = (const float*)d_in[i];

  // workspace carve (4 ping-pong N x 512 buffers + small scratch)
  char* wsb = (char*)d_ws;
  size_t off = 0;
  auto carve = [&](size_t bytes) -> float* {
    float* p = (float*)(wsb + off);
    off += (bytes + 255) & ~(size_t)255;
    return p;
  };
  const size_t bigB = (size_t)NN * HD3 * sizeof(float);
  float* B0   = carve(bigB);
  float* B1   = carve(bigB);
  float* B2   = carve(bigB);
  float* B3   = carve(bigB);
  float* deg  = carve((size_t)NN * sizeof(float));
  float* dinv = carve((size_t)NN * sizeof(float));
  float* sums = carve((size_t)GG * HD3 * sizeof(float));
  float* cnt  = carve((size_t)GG * sizeof(float));
  float* hg   = carve((size_t)GG * HD3 * sizeof(float));
  float* z1   = carve((size_t)GG * HD2 * sizeof(float));
  float* z2   = carve((size_t)GG * HD1 * sizeof(float));

  dim3 blk(256);
  auto g1d = [](long long n) { return dim3((unsigned)((n + 255) / 256)); };

  auto gemm = [&](const float* X_, const float* W_, const float* b_, float* Y_,
                  int n, int k, int m, int relu) {
    dim3 grid((n + T_ROWS - 1) / T_ROWS, (m + T_COLS - 1) / T_COLS);
    gemm_wmma_f32<<<grid, blk, 0, stream>>>(X_, W_, b_, Y_, n, k, m, relu);
  };

  auto leconv = [&](const float* hin, int fin, int fout,
                    const float* W1, const float* b1, const float* W2,
                    const float* W3, const float* b3,
                    float* Aa, float* Bb, float* Agg) {
    dim3 grid((NN + T_ROWS - 1) / T_ROWS, (fout + T_COLS - 1) / T_COLS);
    gemm3_wmma_f32<<<grid, blk, 0, stream>>>(hin, W1, b1, W2, W3, b3,
                                             Aa, Bb, Agg, NN, fin, fout);
    long long tot = (long long)EE * (fout / 4);
    le_edge_k<<<g1d(tot), blk, 0, stream>>>(Aa, Bb, ew, src, dst, Agg, EE, fout);
    long long nt = (long long)NN * fout;
    relu_k<<<g1d(nt), blk, 0, stream>>>(Agg, nt);
  };

  auto gcn = [&](const float* hin, int f, const float* W, const float* b,
                 float* XW, float* Out) {
    gemm(hin, W, nullptr, XW, NN, f, f, 0);
    long long nt = (long long)NN * f;
    gcn_init_k<<<g1d(nt), blk, 0, stream>>>(XW, dinv, b, Out, NN, f);
    long long tot = (long long)EE * (f / 4);
    gcn_edge_k<<<g1d(tot), blk, 0, stream>>>(XW, dinv, ew, src, dst, Out, EE, f);
    relu_k<<<g1d(nt), blk, 0, stream>>>(Out, nt);
  };

  // degree / d^-1/2 once (edge weights identical for all GCN layers)
  set_val_k<<<g1d(NN), blk, 0, stream>>>(deg, NN, 1.0f);   // self-loop weight
  deg_edge_k<<<g1d(EE), blk, 0, stream>>>(deg, dst, ew, EE);
  dinv_k<<<g1d(NN), blk, 0, stream>>>(deg, dinv, NN);

  // stage 1: 128 -> 128
  leconv(x, FIN, HD1, P[4], P[5], P[6], P[7], P[8], B0, B1, B2);
  gcn(B2, HD1, P[9], P[10], B0, B1);
  // stage 2: 128 -> 256
  leconv(B1, HD1, HD2, P[11], P[12], P[13], P[14], P[15], B0, B2, B3);
  gcn(B3, HD2, P[16], P[17], B0, B1);
  // stage 3: 256 -> 512
  leconv(B1, HD2, HD3, P[18], P[19], P[20], P[21], P[22], B0, B2, B3);
  gcn(B3, HD3, P[23], P[24], B0, B2);

  // global mean pool
  set_val_k<<<g1d((long long)GG * HD3), blk, 0, stream>>>(sums, (long long)GG * HD3, 0.0f);
  set_val_k<<<g1d(GG), blk, 0, stream>>>(cnt, GG, 0.0f);
  pool_acc_k<<<g1d((long long)NN * (HD3 / 4)), blk, 0, stream>>>(B2, bat, sums, NN, HD3);
  pool_cnt_k<<<g1d(NN), blk, 0, stream>>>(bat, cnt, NN);
  pool_div_k<<<g1d((long long)GG * HD3), blk, 0, stream>>>(sums, cnt, hg, GG, HD3);

  // MLP head (reuses WMMA GEMM; final layer writes straight to d_out)
  gemm(hg, P[25], P[26], z1, GG, HD3, HD2, 1);
  gemm(z1, P[27], P[28], z2, GG, HD2, HD1, 1);
  gemm(z2, P[29], P[30], (float*)d_out, GG, HD1, 1, 0);
}